// _TinyHybridXG_72722386256531
// MI455X (gfx1250) — compile-verified
//
#include <hip/hip_runtime.h>

typedef float v2f __attribute__((ext_vector_type(2)));
typedef float v8f __attribute__((ext_vector_type(8)));

static __host__ __device__ inline int cdiv_i(long long a, long long b) { return (int)((a + b - 1) / b); }

// ---------------- fill ----------------
__global__ void k_fill(float* __restrict__ p, size_t n, float v) {
    size_t i = (size_t)blockIdx.x * blockDim.x + threadIdx.x;
    if (i < n) p[i] = v;
}

// ---------------- degree: deg[col[e]] += 1 ----------------
__global__ void k_deg(const int* __restrict__ col, float* __restrict__ deg, int nE) {
    int e = blockIdx.x * blockDim.x + threadIdx.x;
    if (e < nE) atomicAdd(&deg[col[e]], 1.0f);
}

// ---------------- deg -> rsqrt(deg) in place ----------------
__global__ void k_rsqrt(float* __restrict__ d, int n) {
    int i = blockIdx.x * blockDim.x + threadIdx.x;
    if (i < n) d[i] = rsqrtf(d[i]);
}

// ---------------- t1 = x @ W1   ([N,9] @ [9,16]) ----------------
__global__ void k_xw1(const float* __restrict__ x, const float* __restrict__ W1,
                      float* __restrict__ t1, int n) {
    int i = blockIdx.x * blockDim.x + threadIdx.x;
    if (i >= n) return;
    float xv[9];
#pragma unroll
    for (int k = 0; k < 9; ++k) xv[k] = x[(size_t)i * 9 + k];
#pragma unroll
    for (int f = 0; f < 16; ++f) {
        float s = 0.0f;
#pragma unroll
        for (int k = 0; k < 9; ++k) s += xv[k] * W1[k * 16 + f];
        t1[(size_t)i * 16 + f] = s;
    }
}

// ---------------- edge message pass: agg[col] += t[row] * dinv[row]*dinv[col] ----------------
// 4 lanes per edge, float4 per lane (cooperative 64B gather of one feature row).
__global__ void k_edge(const float* __restrict__ t, const float* __restrict__ dinv,
                       const int* __restrict__ row, const int* __restrict__ col,
                       float* __restrict__ agg, int nE) {
    long long tid = (long long)blockIdx.x * blockDim.x + threadIdx.x;
    int e = (int)(tid >> 2);
    int g = (int)(tid & 3);
    if (e >= nE) return;
    int r = row[e];
    int c = col[e];
    float norm = dinv[r] * dinv[c];
    const float4 v = *(const float4*)(t + (size_t)r * 16 + g * 4);
    float* dst = agg + (size_t)c * 16 + g * 4;
    atomicAdd(dst + 0, v.x * norm);
    atomicAdd(dst + 1, v.y * norm);
    atomicAdd(dst + 2, v.z * norm);
    atomicAdd(dst + 3, v.w * norm);
}

// ---------------- epilogue: h = relu(agg + t*dinv^2 + b), h written over agg in place ----------
__global__ void k_post(float* __restrict__ agg, const float* __restrict__ t,
                       const float* __restrict__ dinv, const float* __restrict__ b, int n) {
    long long tid = (long long)blockIdx.x * blockDim.x + threadIdx.x;
    int i = (int)(tid >> 2);
    int g = (int)(tid & 3);
    if (i >= n) return;
    float di = dinv[i];
    float self_n = di * di;
    size_t base = (size_t)i * 16 + g * 4;
    float4 a = *(const float4*)(agg + base);
    float4 tv = *(const float4*)(t + base);
    float4 h;
    h.x = fmaxf(a.x + tv.x * self_n + b[g * 4 + 0], 0.0f);
    h.y = fmaxf(a.y + tv.y * self_n + b[g * 4 + 1], 0.0f);
    h.z = fmaxf(a.z + tv.z * self_n + b[g * 4 + 2], 0.0f);
    h.w = fmaxf(a.w + tv.w * self_n + b[g * 4 + 3], 0.0f);
    *(float4*)(agg + base) = h;
}

// ---------------- WMMA: D[tile] = A[tile](16x16) @ B(16x16), f32, K split into 4x4 ---------
// One wave per 16-node tile; EXEC stays all-ones (wave-uniform guard only).
__global__ void k_gemm16_wmma(const float* __restrict__ A, const float* __restrict__ B,
                              float* __restrict__ D, int ntiles) {
    int wave = (int)(((long long)blockIdx.x * blockDim.x + threadIdx.x) >> 5);
    int lane = threadIdx.x & 31;
    if (wave >= ntiles) return;                 // uniform per wave
    const int m  = lane & 15;                   // A row / B col / D col
    const int kh = (lane >> 4) << 1;            // K sub-offset: 0 (lanes 0-15) or 2 (lanes 16-31)
    const float* Ab = A + (size_t)wave * 256;
    v8f acc = {};
#pragma unroll
    for (int s = 0; s < 4; ++s) {
        int k0 = 4 * s + kh;
        v2f a, b;
        a.x = Ab[m * 16 + k0];
        a.y = Ab[m * 16 + k0 + 1];
        b.x = B[k0 * 16 + m];
        b.y = B[(k0 + 1) * 16 + m];
        acc = __builtin_amdgcn_wmma_f32_16x16x4_f32(false, a, false, b, (short)0, acc,
                                                    false, false);
    }
    float* Db = D + (size_t)wave * 256;
    const int mh = (lane >> 4) << 3;            // D row offset: 0 or 8
#pragma unroll
    for (int r = 0; r < 8; ++r) Db[(r + mh) * 16 + m] = acc[r];
}

// scalar tail for N % 16 (not hit for N=500000, kept for generality)
__global__ void k_gemm16_tail(const float* __restrict__ A, const float* __restrict__ B,
                              float* __restrict__ D, int start, int n) {
    long long tid = (long long)blockIdx.x * blockDim.x + threadIdx.x;
    int i = start + (int)(tid >> 4);
    int f = (int)(tid & 15);
    if (i >= n) return;
    float s = 0.0f;
#pragma unroll
    for (int k = 0; k < 16; ++k) s += A[(size_t)i * 16 + k] * B[k * 16 + f];
    D[(size_t)i * 16 + f] = s;
}

// ---------------- mean pool accumulation ----------------
__global__ void k_pool(const float* __restrict__ h, const int* __restrict__ batch,
                       float* __restrict__ gsum, float* __restrict__ gcnt, int n) {
    long long tid = (long long)blockIdx.x * blockDim.x + threadIdx.x;
    int i = (int)(tid >> 2);
    int g = (int)(tid & 3);
    if (i >= n) return;
    int b = batch[i];
    const float4 v = *(const float4*)(h + (size_t)i * 16 + g * 4);
    float* dst = gsum + (size_t)b * 16 + g * 4;
    atomicAdd(dst + 0, v.x);
    atomicAdd(dst + 1, v.y);
    atomicAdd(dst + 2, v.z);
    atomicAdd(dst + 3, v.w);
    if (g == 0) atomicAdd(&gcnt[b], 1.0f);
}

// ---------------- per-graph head MLP: out = relu(concat(emb,meta)@Wh1+bh1)@Wh2+bh2 -------
__global__ void k_head(const float* __restrict__ gsum, const float* __restrict__ gcnt,
                       const float* __restrict__ meta, const float* __restrict__ Wh1,
                       const float* __restrict__ bh1, const float* __restrict__ Wh2,
                       const float* __restrict__ bh2, float* __restrict__ out, int G) {
    int g = blockIdx.x * blockDim.x + threadIdx.x;
    if (g >= G) return;
    float inv = 1.0f / fmaxf(gcnt[g], 1.0f);
    float emb[16];
#pragma unroll
    for (int f = 0; f < 16; ++f) emb[f] = gsum[(size_t)g * 16 + f] * inv;
    float o = bh2[0];
    for (int j = 0; j < 16; ++j) {
        float s = bh1[j];
#pragma unroll
        for (int f = 0; f < 16; ++f) s += emb[f] * Wh1[f * 16 + j];
        for (int f = 0; f < 27; ++f) s += meta[(size_t)g * 27 + f] * Wh1[(16 + f) * 16 + j];
        o += fmaxf(s, 0.0f) * Wh2[j];
    }
    out[g] = o;
}

extern "C" void kernel_launch(void* const* d_in, const int* in_sizes, int n_in,
                              void* d_out, int out_size, void* d_ws, size_t ws_size,
                              hipStream_t stream) {
    const float* x    = (const float*)d_in[0];
    const int*   ei   = (const int*)d_in[1];     // [2, E] int32
    const int*   bat  = (const int*)d_in[2];     // [N]
    const float* meta = (const float*)d_in[3];   // [G, 27]
    const float* W1   = (const float*)d_in[4];
    const float* b1   = (const float*)d_in[5];
    const float* W2   = (const float*)d_in[6];
    const float* b2   = (const float*)d_in[7];
    const float* Wh1  = (const float*)d_in[8];
    const float* bh1  = (const float*)d_in[9];
    const float* Wh2  = (const float*)d_in[10];
    const float* bh2  = (const float*)d_in[11];
    float* out = (float*)d_out;

    const int N = in_sizes[0] / 9;
    const int E = in_sizes[1] / 2;
    const int G = in_sizes[3] / 27;
    const int* row = ei;
    const int* col = ei + E;

    // workspace layout (floats): dinv[N] | bufT[N*16] | bufAgg[N*16] | gsum[G*16] | gcnt[G]
    float* dinv   = (float*)d_ws;
    float* bufT   = dinv + N;
    float* bufAgg = bufT + (size_t)N * 16;
    float* gsum   = bufAgg + (size_t)N * 16;
    float* gcnt   = gsum + (size_t)G * 16;

    const int BT = 256;
    // init: deg starts at 1.0 (self-loop), agg=0, pool=0 (gsum+gcnt contiguous)
    k_fill<<<cdiv_i(N, BT), BT, 0, stream>>>(dinv, (size_t)N, 1.0f);
    k_fill<<<cdiv_i((long long)N * 16, BT), BT, 0, stream>>>(bufAgg, (size_t)N * 16, 0.0f);
    k_fill<<<cdiv_i((long long)G * 17, BT), BT, 0, stream>>>(gsum, (size_t)G * 17, 0.0f);

    // degrees and symmetric normalization
    k_deg<<<cdiv_i(E, BT), BT, 0, stream>>>(col, dinv, E);
    k_rsqrt<<<cdiv_i(N, BT), BT, 0, stream>>>(dinv, N);

    // ---- layer 1 ----
    k_xw1<<<cdiv_i(N, BT), BT, 0, stream>>>(x, W1, bufT, N);
    k_edge<<<cdiv_i((long long)E * 4, BT), BT, 0, stream>>>(bufT, dinv, row, col, bufAgg, E);
    k_post<<<cdiv_i((long long)N * 4, BT), BT, 0, stream>>>(bufAgg, bufT, dinv, b1, N); // h1 in bufAgg

    // ---- layer 2 linear via WMMA: bufT = h1 @ W2 ----
    int ntiles = N / 16;
    if (ntiles > 0)
        k_gemm16_wmma<<<cdiv_i((long long)ntiles * 32, BT), BT, 0, stream>>>(bufAgg, W2, bufT, ntiles);
    int rem = N - ntiles * 16;
    if (rem > 0)
        k_gemm16_tail<<<cdiv_i((long long)rem * 16, BT), BT, 0, stream>>>(bufAgg, W2, bufT, ntiles * 16, N);

    // ---- layer 2 aggregation ----
    k_fill<<<cdiv_i((long long)N * 16, BT), BT, 0, stream>>>(bufAgg, (size_t)N * 16, 0.0f);
    k_edge<<<cdiv_i((long long)E * 4, BT), BT, 0, stream>>>(bufT, dinv, row, col, bufAgg, E);
    k_post<<<cdiv_i((long long)N * 4, BT), BT, 0, stream>>>(bufAgg, bufT, dinv, b2, N); // h2 in bufAgg

    // ---- pooling + head ----
    k_pool<<<cdiv_i((long long)N * 4, BT), BT, 0, stream>>>(bufAgg, bat, gsum, gcnt, N);
    k_head<<<cdiv_i(G, BT), BT, 0, stream>>>(gsum, gcnt, meta, Wh1, bh1, Wh2, bh2, out, G);
}